// GCN_2972117368897
// MI455X (gfx1250) — compile-verified
//
#include <hip/hip_runtime.h>
#include <hip/hip_bf16.h>

typedef float v2f __attribute__((ext_vector_type(2)));
typedef float v8f __attribute__((ext_vector_type(8)));

// ---------------------------------------------------------------------------
// Degree / normalization
// ---------------------------------------------------------------------------
__global__ void deg_init_kernel(float* __restrict__ deg, int n) {
    int i = blockIdx.x * blockDim.x + threadIdx.x;
    if (i < n) deg[i] = 1.0f;   // self-loop contribution
}

__global__ void deg_accum_kernel(const int* __restrict__ dst, float* __restrict__ deg, int e) {
    int i = blockIdx.x * blockDim.x + threadIdx.x;
    if (i < e) atomicAdd(&deg[dst[i]], 1.0f);
}

__global__ void rsqrt_kernel(float* __restrict__ deg, int n) {
    int i = blockIdx.x * blockDim.x + threadIdx.x;
    if (i < n) deg[i] = rsqrtf(deg[i]);
}

// ---------------------------------------------------------------------------
// f32 GEMM on the WMMA pipe: C[M,N] = A[M,K] * B[K,N]
// One wave per 16x16 output tile, V_WMMA_F32_16X16X4_F32 over K.
// Requires M%16==0, N%16==0, K%4==0 (holds: 50000=3125*16, K=128/64, N=64/32).
// ---------------------------------------------------------------------------
__global__ void gemm_wmma_f32_kernel(const float* __restrict__ A,
                                     const float* __restrict__ B,
                                     float* __restrict__ C,
                                     int M, int K, int N) {
    const int lane  = threadIdx.x & 31;
    const int wave  = threadIdx.x >> 5;
    const int tilesN = N >> 4;
    const int total  = (M >> 4) * tilesN;
    const int tile   = blockIdx.x * (blockDim.x >> 5) + wave;
    if (tile >= total) return;          // wave-uniform: EXEC stays all-ones inside

    const int mt   = tile / tilesN;
    const int nt   = tile - mt * tilesN;
    const int half = lane >> 4;         // 0: lanes 0-15, 1: lanes 16-31
    const int l15  = lane & 15;

    // A layout (16x4 f32): lane holds row m=l15; VGPR0 = K+2*half, VGPR1 = K+2*half+1
    const float* arow = A + (long long)(mt * 16 + l15) * K;
    // B layout (4x16 f32): lane holds col n=l15; VGPR r = row (k + r), halves split K by +2
    const float* bcol = B + nt * 16 + l15;

    v8f c = {};
    for (int k = 0; k < K; k += 4) {
        v2f a, b;
        a.x = arow[k + 2 * half];
        a.y = arow[k + 2 * half + 1];
        b.x = bcol[(long long)(k + 2 * half) * N];
        b.y = bcol[(long long)(k + 2 * half + 1) * N];
        c = __builtin_amdgcn_wmma_f32_16x16x4_f32(
                /*neg_a=*/false, a, /*neg_b=*/false, b,
                /*c_mod=*/(short)0, c, /*reuse_a=*/false, /*reuse_b=*/false);
    }

    // C/D layout: VGPR r -> row r + 8*half, col = l15
    float* crow = C + (long long)(mt * 16) * N + nt * 16 + l15;
#pragma unroll
    for (int r = 0; r < 8; ++r)
        crow[(long long)(r + 8 * half) * N] = c[r];
}

// ---------------------------------------------------------------------------
// Aggregation: out starts with self-loop term + bias, then edge scatter-adds.
// ---------------------------------------------------------------------------
__global__ void agg_init_kernel(const float* __restrict__ xw,
                                const float* __restrict__ dinv,
                                const float* __restrict__ bias,
                                float* __restrict__ out,
                                int n, int F) {
    long long i = (long long)blockIdx.x * blockDim.x + threadIdx.x;
    long long total = (long long)n * F;
    if (i >= total) return;
    int node = (int)(i / F);
    int f    = (int)(i - (long long)node * F);
    float di = dinv[node];
    out[i] = xw[i] * di * di + bias[f];
}

__global__ void edge_scatter_kernel(const int* __restrict__ src,
                                    const int* __restrict__ dst,
                                    const float* __restrict__ dinv,
                                    const float* __restrict__ xw,
                                    float* __restrict__ out,
                                    int E, int F) {
    long long t = (long long)blockIdx.x * blockDim.x + threadIdx.x;
    long long total = (long long)E * F;
    if (t >= total) return;
    int e = (int)(t / F);
    int f = (int)(t - (long long)e * F);
    int s = src[e];
    int d = dst[e];
    float nrm = dinv[s] * dinv[d];
    atomicAdd(&out[(long long)d * F + f], xw[(long long)s * F + f] * nrm);
}

__global__ void relu_kernel(float* __restrict__ x, long long n) {
    long long i = (long long)blockIdx.x * blockDim.x + threadIdx.x;
    if (i < n) x[i] = fmaxf(x[i], 0.0f);
}

// ---------------------------------------------------------------------------
// Launcher
// ---------------------------------------------------------------------------
extern "C" void kernel_launch(void* const* d_in, const int* in_sizes, int n_in,
                              void* d_out, int out_size, void* d_ws, size_t ws_size,
                              hipStream_t stream) {
    const float* x   = (const float*)d_in[0];
    const int*   ei  = (const int*)d_in[1];
    const float* W1  = (const float*)d_in[2];
    const float* b1  = (const float*)d_in[3];
    const float* W2  = (const float*)d_in[4];
    const float* b2  = (const float*)d_in[5];
    float*       out = (float*)d_out;

    const int E     = in_sizes[1] / 2;             // 800000
    const int F_HID = in_sizes[3];                 // 64
    const int F_IN  = in_sizes[2] / F_HID;         // 128
    const int F_OUT = in_sizes[5];                 // 32
    const int N     = in_sizes[0] / F_IN;          // 50000

    const int* src = ei;
    const int* dst = ei + E;

    // Workspace layout (floats)
    float* deg = (float*)d_ws;                     // N (also dinv, in place)
    float* xw1 = deg + 65536;                      // N * F_HID
    float* h   = xw1 + (long long)N * F_HID;       // N * F_HID (agg buffer, relu in place)
    float* hw2 = h   + (long long)N * F_HID;       // N * F_OUT

    const int TB = 256;

    // 1) degrees -> dinv
    deg_init_kernel<<<(N + TB - 1) / TB, TB, 0, stream>>>(deg, N);
    deg_accum_kernel<<<(E + TB - 1) / TB, TB, 0, stream>>>(dst, deg, E);
    rsqrt_kernel<<<(N + TB - 1) / TB, TB, 0, stream>>>(deg, N);

    // 2) xw1 = x @ W1  (WMMA f32)
    {
        int tiles = (N / 16) * (F_HID / 16);
        int wavesPerBlock = TB / 32;
        int blocks = (tiles + wavesPerBlock - 1) / wavesPerBlock;
        gemm_wmma_f32_kernel<<<blocks, TB, 0, stream>>>(x, W1, xw1, N, F_IN, F_HID);
    }

    // 3) layer-1 aggregation into h, then ReLU
    {
        long long total = (long long)N * F_HID;
        agg_init_kernel<<<(int)((total + TB - 1) / TB), TB, 0, stream>>>(xw1, deg, b1, h, N, F_HID);
        long long etotal = (long long)E * F_HID;
        edge_scatter_kernel<<<(int)((etotal + TB - 1) / TB), TB, 0, stream>>>(src, dst, deg, xw1, h, E, F_HID);
        relu_kernel<<<(int)((total + TB - 1) / TB), TB, 0, stream>>>(h, total);
    }

    // 4) hw2 = h @ W2  (WMMA f32)
    {
        int tiles = (N / 16) * (F_OUT / 16);
        int wavesPerBlock = TB / 32;
        int blocks = (tiles + wavesPerBlock - 1) / wavesPerBlock;
        gemm_wmma_f32_kernel<<<blocks, TB, 0, stream>>>(h, W2, hw2, N, F_HID, F_OUT);
    }

    // 5) layer-2 aggregation straight into d_out (init also adds bias)
    {
        long long total = (long long)N * F_OUT;
        agg_init_kernel<<<(int)((total + TB - 1) / TB), TB, 0, stream>>>(hw2, deg, b2, out, N, F_OUT);
        long long etotal = (long long)E * F_OUT;
        edge_scatter_kernel<<<(int)((etotal + TB - 1) / TB), TB, 0, stream>>>(src, dst, deg, hw2, out, E, F_OUT);
    }
}